// MyLoss_24773371363366
// MI455X (gfx1250) — compile-verified
//
#include <hip/hip_runtime.h>
#include <hip/hip_bf16.h>

typedef float v2f __attribute__((ext_vector_type(2)));
typedef float v8f __attribute__((ext_vector_type(8)));

#define N_PTS 8192
#define DIM   128
#define MT    128               // macro tile edge
#define NB    (N_PTS / MT)      // 64 blocks per dim
#define LDSW  132               // padded LDS row stride (dwords), 132 % 64 == 4 -> conflict-free b64 frag loads

struct Partial { float ss, sd; unsigned cs, cd; };

// ---------------- phase 1: row squared norms ----------------
__global__ void sqnorm_kernel(const float* __restrict__ pred, float* __restrict__ sq) {
  int wave = threadIdx.x >> 5;
  int lane = threadIdx.x & 31;
  int row  = blockIdx.x * 8 + wave;
  const float4* p = (const float4*)(pred + (size_t)row * DIM);
  float4 v = p[lane];                       // 32 lanes * 4 floats = 128 = DIM
  float s = v.x * v.x + v.y * v.y + v.z * v.z + v.w * v.w;
#pragma unroll
  for (int off = 16; off > 0; off >>= 1) s += __shfl_xor(s, off, 32);
  if (lane == 0) sq[row] = s;
}

// ---------------- phase 2: WMMA pair tiles ----------------
__global__ void __launch_bounds__(256) pair_tile_kernel(
    const float* __restrict__ pred, const int* __restrict__ tgt,
    const float* __restrict__ sq, Partial* __restrict__ partials) {
  __shared__ float lds[2 * MT * LDSW];      // Pi panel then Pj panel (~132 KB)

  const int bj = blockIdx.x, bi = blockIdx.y;
  const int tid = threadIdx.x;
  Partial* out = &partials[bi * NB + bj];
  if (bi > bj) {                            // lower-triangle macro blocks do nothing
    if (tid == 0) { out->ss = 0.f; out->sd = 0.f; out->cs = 0u; out->cd = 0u; }
    return;
  }

  float* Pi = lds;
  float* Pj = lds + MT * LDSW;
  // stage both 128x128 panels, coalesced float4 loads (one full 512B row per wave)
  for (int idx = tid; idx < MT * (DIM / 4); idx += 256) {
    int row = idx >> 5;                     // 32 float4 per row
    int c4  = idx & 31;
    float4 vi = *(const float4*)(pred + (size_t)(bi * MT + row) * DIM + c4 * 4);
    float4 vj = *(const float4*)(pred + (size_t)(bj * MT + row) * DIM + c4 * 4);
    *(float4*)&Pi[row * LDSW + c4 * 4] = vi;
    *(float4*)&Pj[row * LDSW + c4 * 4] = vj;
  }
  __syncthreads();

  const int w    = tid >> 5;                // wave id: i sub-block
  const int lane = tid & 31;
  const int half = lane >> 4;               // K half for A/B fragments
  const int m    = lane & 15;               // M (for A) / N (for B,C,D)

  v8f acc[8] = {};                          // 8 j-tiles of 16x16 f32

  const float* arow = &Pi[(16 * w + m) * LDSW + 2 * half];
  const float* brow = &Pj[m * LDSW + 2 * half];

#pragma unroll 4
  for (int K = 0; K < DIM; K += 4) {
    v2f a = *(const v2f*)(arow + K);        // A frag: rows 16w..16w+15, K..K+3
#pragma unroll
    for (int j = 0; j < 8; ++j) {
      v2f b = *(const v2f*)(brow + j * 16 * LDSW + K);
      acc[j] = __builtin_amdgcn_wmma_f32_16x16x4_f32(
          false, a, false, b, (short)0, acc[j], false, false);
    }
  }

  // ---- epilogue: d2 = sq_i + sq_j - 2*gram ; mask i<j ; accumulate ----
  const int gi0 = bi * MT + 16 * w + 8 * half;   // gi = gi0 + r  (C/D: M = r + 8*half)
  float sqi[8]; int ti[8];
#pragma unroll
  for (int r = 0; r < 8; ++r) { sqi[r] = sq[gi0 + r]; ti[r] = tgt[gi0 + r]; }

  float ssum = 0.f, dsum = 0.f;
  unsigned csame = 0u, cdiff = 0u;
#pragma unroll
  for (int j = 0; j < 8; ++j) {
    int   gj  = bj * MT + 16 * j + m;            // C/D: N = m
    float sqj = sq[gj];
    int   tj  = tgt[gj];
#pragma unroll
    for (int r = 0; r < 8; ++r) {
      int gi = gi0 + r;
      if (gi < gj) {
        float g  = acc[j][r];
        float d2 = fmaxf(sqi[r] + sqj - 2.0f * g, 0.0f);
        float d  = sqrtf(d2);
        if (ti[r] == tj) { ssum += d;        csame++; }
        else             { dsum += expf(-d); cdiff++; }
      }
    }
  }

  // ---- deterministic workgroup reduction (reuse LDS) ----
  __syncthreads();
  float*    rs = lds;                        // [0..255]=ssum, [256..511]=dsum
  unsigned* ru = (unsigned*)(lds + 512);     // [0..255]=csame, [256..511]=cdiff
  rs[tid] = ssum;  rs[256 + tid] = dsum;
  ru[tid] = csame; ru[256 + tid] = cdiff;
  __syncthreads();
  for (int s = 128; s > 0; s >>= 1) {
    if (tid < s) {
      rs[tid]       += rs[tid + s];
      rs[256 + tid] += rs[256 + tid + s];
      ru[tid]       += ru[tid + s];
      ru[256 + tid] += ru[256 + tid + s];
    }
    __syncthreads();
  }
  if (tid == 0) { out->ss = rs[0]; out->sd = rs[256]; out->cs = ru[0]; out->cd = ru[256]; }
}

// ---------------- phase 3: final reduce + loss ----------------
__global__ void finalize_kernel(const Partial* __restrict__ partials, float* __restrict__ out) {
  __shared__ double   rs[512];
  __shared__ unsigned ru[512];
  int tid = threadIdx.x;
  double ss = 0.0, sd = 0.0;
  unsigned cs = 0u, cd = 0u;
  for (int i = tid; i < NB * NB; i += 256) {
    ss += (double)partials[i].ss;
    sd += (double)partials[i].sd;
    cs += partials[i].cs;
    cd += partials[i].cd;
  }
  rs[tid] = ss; rs[256 + tid] = sd;
  ru[tid] = cs; ru[256 + tid] = cd;
  __syncthreads();
  for (int s = 128; s > 0; s >>= 1) {
    if (tid < s) {
      rs[tid]       += rs[tid + s];
      rs[256 + tid] += rs[256 + tid + s];
      ru[tid]       += ru[tid + s];
      ru[256 + tid] += ru[256 + tid + s];
    }
    __syncthreads();
  }
  if (tid == 0) {
    double same_den = (double)(ru[0]   > 0u ? ru[0]   : 1u);
    double diff_den = (double)(ru[256] > 0u ? ru[256] : 1u);
    double same_loss = rs[0]   / same_den;
    double diff_loss = rs[256] / diff_den;
    out[0] = (float)(1.0 * (0.01 * same_loss + 100.0 * diff_loss));
  }
}

extern "C" void kernel_launch(void* const* d_in, const int* in_sizes, int n_in,
                              void* d_out, int out_size, void* d_ws, size_t ws_size,
                              hipStream_t stream) {
  (void)in_sizes; (void)n_in; (void)out_size; (void)ws_size;
  const float* pred = (const float*)d_in[0];
  const int*   tgt  = (const int*)d_in[1];

  float*   sq       = (float*)d_ws;                                  // 8192 floats
  Partial* partials = (Partial*)((char*)d_ws + N_PTS * sizeof(float)); // 4096 Partials

  sqnorm_kernel<<<N_PTS / 8, 256, 0, stream>>>(pred, sq);
  dim3 grid(NB, NB);
  pair_tile_kernel<<<grid, 256, 0, stream>>>(pred, tgt, sq, partials);
  finalize_kernel<<<1, 256, 0, stream>>>(partials, (float*)d_out);
}